// GNNRecommender_3092376453605
// MI455X (gfx1250) — compile-verified
//
#include <hip/hip_runtime.h>
#include <cstdint>
#include <cstddef>

typedef __attribute__((ext_vector_type(2))) float v2f;
typedef __attribute__((ext_vector_type(8))) float v8f;

// ---------- helpers ----------

// Order-preserving float -> u32 encoding (for atomicMax-based segment max)
__device__ __forceinline__ unsigned enc_f32(float f) {
  unsigned b = __float_as_uint(f);
  return (b & 0x80000000u) ? ~b : (b | 0x80000000u);
}
__device__ __forceinline__ float dec_f32(unsigned u) {
  unsigned b = (u & 0x80000000u) ? (u & 0x7FFFFFFFu) : ~u;
  return __uint_as_float(b);
}

__device__ __forceinline__ float leaky02(float v) { return v > 0.0f ? v : 0.2f * v; }

// ---------- utility kernels ----------

__global__ void zero_f32_kernel(float* __restrict__ p, size_t n) {
  size_t i = (size_t)blockIdx.x * blockDim.x + threadIdx.x;
  if (i < n) p[i] = 0.0f;
}

__global__ void fill_u32_kernel(unsigned* __restrict__ p, size_t n, unsigned v) {
  size_t i = (size_t)blockIdx.x * blockDim.x + threadIdx.x;
  if (i < n) p[i] = v;
}

// ---------- fp32 WMMA GEMM: out[M,Nc] = A[M,K] @ W[K,Nc] ----------
// One wave computes a 16x64 output strip using V_WMMA_F32_16X16X4_F32.
// Requires: M % 16 == 0, K % 4 == 0, Nc % 64 == 0 (true for all layers here).
__global__ void gemm_wmma_f32(const float* __restrict__ A,
                              const float* __restrict__ W,
                              float* __restrict__ C,
                              int M, int K, int Nc) {
  int wave = (int)((blockIdx.x * (size_t)blockDim.x + threadIdx.x) >> 5);
  int lane = threadIdx.x & 31;
  int ntiles = Nc >> 6;            // 64-wide strips
  int mtiles = M >> 4;
  if (wave >= mtiles * ntiles) return;   // wave-uniform exit, EXEC stays full
  int mb = (wave / ntiles) << 4;
  int nb = (wave % ntiles) << 6;

  int m     = lane & 15;
  int kb    = (lane >> 4) << 1;    // 0 or 2
  int nlane = lane & 15;

  v8f acc0 = {}, acc1 = {}, acc2 = {}, acc3 = {};

  for (int k = 0; k < K; k += 4) {
    // A fragment: lane holds A[mb+m, k+kb] and A[mb+m, k+kb+1] (8B aligned)
    v2f a = *(const v2f*)(A + (size_t)(mb + m) * K + k + kb);
    const float* w0 = W + (size_t)(k + kb) * Nc + nb + nlane;
    const float* w1 = W + (size_t)(k + kb + 1) * Nc + nb + nlane;
    v2f b0, b1, b2, b3;
    b0.x = w0[0];  b0.y = w1[0];
    b1.x = w0[16]; b1.y = w1[16];
    b2.x = w0[32]; b2.y = w1[32];
    b3.x = w0[48]; b3.y = w1[48];
    acc0 = __builtin_amdgcn_wmma_f32_16x16x4_f32(false, a, false, b0, (short)0, acc0, false, false);
    acc1 = __builtin_amdgcn_wmma_f32_16x16x4_f32(false, a, false, b1, (short)0, acc1, false, false);
    acc2 = __builtin_amdgcn_wmma_f32_16x16x4_f32(false, a, false, b2, (short)0, acc2, false, false);
    acc3 = __builtin_amdgcn_wmma_f32_16x16x4_f32(false, a, false, b3, (short)0, acc3, false, false);
  }

  // C/D layout: VGPR r -> row r (lanes 0-15) and row r+8 (lanes 16-31)
  int rowhi = (lane >> 4) << 3;
  #pragma unroll
  for (int r = 0; r < 8; ++r) {
    float* crow = C + (size_t)(mb + r + rowhi) * Nc + nb + nlane;
    crow[0]  = acc0[r];
    crow[16] = acc1[r];
    crow[32] = acc2[r];
    crow[48] = acc3[r];
  }
}

// ---------- attention scores: s[n,h] = sum_c h[n,h,c] * a[h,c] ----------
__global__ void attn_score_kernel(const float* __restrict__ hlin,
                                  const float* __restrict__ a_src,
                                  const float* __restrict__ a_dst,
                                  float* __restrict__ s_src,
                                  float* __restrict__ s_dst,
                                  int N, int H, int Cc) {
  int idx = (int)(blockIdx.x * (size_t)blockDim.x + threadIdx.x);
  if (idx >= N * H) return;
  int h = idx % H, n = idx / H;
  const float* row = hlin + (size_t)n * H * Cc + (size_t)h * Cc;
  const float* as  = a_src + (size_t)h * Cc;
  const float* ad  = a_dst + (size_t)h * Cc;
  float ss = 0.0f, sd = 0.0f;
  for (int c = 0; c < Cc; ++c) {
    float v = row[c];
    ss += v * as[c];
    sd += v * ad[c];
  }
  s_src[idx] = ss;
  s_dst[idx] = sd;
}

// ---------- segment softmax pass 1: per-dst max of leaky_relu scores ----------
__global__ void edge_max_kernel(const int* __restrict__ ei,
                                const float* __restrict__ s_src,
                                const float* __restrict__ s_dst,
                                unsigned* __restrict__ seg_max_u,
                                int E, int N, int H) {
  int idx = (int)(blockIdx.x * (size_t)blockDim.x + threadIdx.x);
  int total = (E + N) * H;
  if (idx >= total) return;
  int h = idx % H, e = idx / H;
  int s, d;
  if (e < E) { s = ei[e]; d = ei[E + e]; } else { s = d = e - E; }
  float v = leaky02(s_src[s * H + h] + s_dst[d * H + h]);
  atomicMax(&seg_max_u[d * H + h], enc_f32(v));
}

// ---------- segment softmax pass 2: per-dst sum of exp(e - max) ----------
__global__ void edge_sum_kernel(const int* __restrict__ ei,
                                const float* __restrict__ s_src,
                                const float* __restrict__ s_dst,
                                const unsigned* __restrict__ seg_max_u,
                                float* __restrict__ seg_sum,
                                int E, int N, int H) {
  int idx = (int)(blockIdx.x * (size_t)blockDim.x + threadIdx.x);
  int total = (E + N) * H;
  if (idx >= total) return;
  int h = idx % H, e = idx / H;
  int s, d;
  if (e < E) { s = ei[e]; d = ei[E + e]; } else { s = d = e - E; }
  float v = leaky02(s_src[s * H + h] + s_dst[d * H + h]);
  float m = dec_f32(seg_max_u[d * H + h]);
  atomicAdd(&seg_sum[d * H + h], __expf(v - m));
}

// ---------- pass 3: agg[dst,h,:] += alpha * h[src,h,:]  (one wave / edge-head) ----------
__global__ void edge_agg_kernel(const int* __restrict__ ei,
                                const float* __restrict__ s_src,
                                const float* __restrict__ s_dst,
                                const unsigned* __restrict__ seg_max_u,
                                const float* __restrict__ seg_sum,
                                const float* __restrict__ hlin,
                                float* __restrict__ agg,
                                int E, int N, int H, int Cc) {
  size_t tid = blockIdx.x * (size_t)blockDim.x + threadIdx.x;
  int wid = (int)(tid >> 5);
  int lane = threadIdx.x & 31;
  int total = (E + N) * H;
  if (wid >= total) return;
  int h = wid % H, e = wid / H;
  int s, d;
  if (e < E) { s = ei[e]; d = ei[E + e]; } else { s = d = e - E; }
  float v = leaky02(s_src[s * H + h] + s_dst[d * H + h]);
  float m = dec_f32(seg_max_u[d * H + h]);
  float alpha = __expf(v - m) / seg_sum[d * H + h];
  const float* hs = hlin + (size_t)s * H * Cc + (size_t)h * Cc;
  float* ad = agg + (size_t)d * H * Cc + (size_t)h * Cc;
  for (int c = lane; c < Cc; c += 32)
    atomicAdd(&ad[c], alpha * hs[c]);
}

// ---------- out = elu(layernorm(in + bias) * g + b)  (one wave / node) ----------
__global__ void ln_elu_kernel(const float* __restrict__ in,
                              const float* __restrict__ bias,
                              const float* __restrict__ g,
                              const float* __restrict__ b,
                              float* __restrict__ out,
                              int N, int D) {
  size_t tid = blockIdx.x * (size_t)blockDim.x + threadIdx.x;
  int node = (int)(tid >> 5);
  int lane = threadIdx.x & 31;
  if (node >= N) return;
  const float* row = in + (size_t)node * D;
  float sum = 0.0f, sumsq = 0.0f;
  for (int c = lane; c < D; c += 32) {
    float v = row[c] + bias[c];
    sum += v;
    sumsq += v * v;
  }
  #pragma unroll
  for (int o = 16; o > 0; o >>= 1) {
    sum   += __shfl_xor(sum, o, 32);
    sumsq += __shfl_xor(sumsq, o, 32);
  }
  float invD = 1.0f / (float)D;
  float mu = sum * invD;
  float var = sumsq * invD - mu * mu;
  float inv = rsqrtf(var + 1e-5f);
  for (int c = lane; c < D; c += 32) {
    float v = row[c] + bias[c];
    float y = (v - mu) * inv * g[c] + b[c];
    out[(size_t)node * D + c] = y > 0.0f ? y : (__expf(y) - 1.0f);
  }
}

// ---------- host-side launcher ----------

static inline unsigned gridFor(size_t threads) {
  return (unsigned)((threads + 255) / 256);
}

extern "C" void kernel_launch(void* const* d_in, const int* in_sizes, int n_in,
                              void* d_out, int out_size, void* d_ws, size_t ws_size,
                              hipStream_t stream) {
  const float* x      = (const float*)d_in[0];
  const int*   ei     = (const int*)d_in[1];
  const float* W1     = (const float*)d_in[2];
  const float* a1s    = (const float*)d_in[3];
  const float* a1d    = (const float*)d_in[4];
  const float* b1     = (const float*)d_in[5];
  const float* ln1g   = (const float*)d_in[6];
  const float* ln1b   = (const float*)d_in[7];
  const float* W2     = (const float*)d_in[8];
  const float* a2s    = (const float*)d_in[9];
  const float* a2d    = (const float*)d_in[10];
  const float* b2     = (const float*)d_in[11];
  const float* ln2g   = (const float*)d_in[12];
  const float* ln2b   = (const float*)d_in[13];
  const float* W3     = (const float*)d_in[14];
  const float* a3s    = (const float*)d_in[15];
  const float* a3d    = (const float*)d_in[16];
  const float* b3     = (const float*)d_in[17];
  const float* ln3g   = (const float*)d_in[18];
  const float* ln3b   = (const float*)d_in[19];
  const float* dW1    = (const float*)d_in[20];
  const float* db1    = (const float*)d_in[21];
  const float* lnd1g  = (const float*)d_in[22];
  const float* lnd1b  = (const float*)d_in[23];
  const float* dW2    = (const float*)d_in[24];
  const float* db2    = (const float*)d_in[25];
  const float* lnd2g  = (const float*)d_in[26];
  const float* lnd2b  = (const float*)d_in[27];

  const int N = in_sizes[0] / 128;   // 50000
  const int E = in_sizes[1] / 2;     // 400000

  // workspace layout
  char* ws = (char*)d_ws;
  size_t big   = (size_t)N * 512 * sizeof(float);
  size_t small = (size_t)N * 4 * sizeof(float);
  float*    bufLin  = (float*)(ws);                       // GEMM output (per-layer h)
  float*    bufAgg  = (float*)(ws + big);                 // scatter accumulator
  float*    bufAct  = (float*)(ws + 2 * big);             // layer activations
  float*    s_src   = (float*)(ws + 3 * big);
  float*    s_dst   = (float*)(ws + 3 * big + small);
  unsigned* seg_max = (unsigned*)(ws + 3 * big + 2 * small);
  float*    seg_sum = (float*)(ws + 3 * big + 3 * small);

  auto gemm = [&](const float* A, const float* W, float* C, int M, int K, int Nc) {
    size_t waves = (size_t)(M / 16) * (Nc / 64);
    gemm_wmma_f32<<<gridFor(waves * 32), 256, 0, stream>>>(A, W, C, M, K, Nc);
  };

  auto gat_layer = [&](const float* in_act, const float* W,
                       const float* as, const float* ad,
                       const float* bias, const float* g, const float* be,
                       int Cin, int H, int Cc, float* out_act) {
    int D = H * Cc;
    gemm(in_act, W, bufLin, N, Cin, D);
    attn_score_kernel<<<gridFor((size_t)N * H), 256, 0, stream>>>(
        bufLin, as, ad, s_src, s_dst, N, H, Cc);
    fill_u32_kernel<<<gridFor((size_t)N * H), 256, 0, stream>>>(seg_max, (size_t)N * H, 0u);
    zero_f32_kernel<<<gridFor((size_t)N * H), 256, 0, stream>>>(seg_sum, (size_t)N * H);
    zero_f32_kernel<<<gridFor((size_t)N * D), 256, 0, stream>>>(bufAgg, (size_t)N * D);
    size_t EH = (size_t)(E + N) * H;
    edge_max_kernel<<<gridFor(EH), 256, 0, stream>>>(ei, s_src, s_dst, seg_max, E, N, H);
    edge_sum_kernel<<<gridFor(EH), 256, 0, stream>>>(ei, s_src, s_dst, seg_max, seg_sum, E, N, H);
    edge_agg_kernel<<<gridFor(EH * 32), 256, 0, stream>>>(
        ei, s_src, s_dst, seg_max, seg_sum, bufLin, bufAgg, E, N, H, Cc);
    ln_elu_kernel<<<gridFor((size_t)N * 32), 256, 0, stream>>>(
        bufAgg, bias, g, be, out_act, N, D);
  };

  // encoder
  gat_layer(x,      W1, a1s, a1d, b1, ln1g, ln1b, 128, 4, 128, bufAct);  // -> [N,512]
  gat_layer(bufAct, W2, a2s, a2d, b2, ln2g, ln2b, 512, 2, 128, bufAct);  // -> [N,256]
  gat_layer(bufAct, W3, a3s, a3d, b3, ln3g, ln3b, 256, 1, 64,  bufAct);  // -> [N,64]

  // decoder
  gemm(bufAct, dW1, bufLin, N, 64, 128);
  ln_elu_kernel<<<gridFor((size_t)N * 32), 256, 0, stream>>>(
      bufLin, db1, lnd1g, lnd1b, bufAgg, N, 128);
  gemm(bufAgg, dW2, bufLin, N, 128, 128);
  ln_elu_kernel<<<gridFor((size_t)N * 32), 256, 0, stream>>>(
      bufLin, db2, lnd2g, lnd2b, (float*)d_out, N, 128);
}